// ESMProjectionHead_4080218931903
// MI455X (gfx1250) — compile-verified
//
#include <hip/hip_runtime.h>

typedef __attribute__((ext_vector_type(2))) float v2f;
typedef __attribute__((ext_vector_type(8))) float v8f;

#define B_N    64
#define L_N    2048
#define D_N    960
#define RL_N   16
#define RD_N   16
#define DLAT_N 512
#define CHUNK  64
#define KSTEPS (D_N / 4) /* 240 wmma_f32_16x16x4 steps per 16-row tile */

// ---------------------------------------------------------------------------
// Kernel 1: fused  U[b] += A_c[chunk]^T * diag(mask) * (emb[b,chunk] @ B_c)
// grid (32 chunks, 64 batch), 128 threads = 4 waves, each wave owns a 16-row
// tile and runs V_WMMA_F32_16X16X4_F32 over K=960.
// ---------------------------------------------------------------------------
__global__ __launch_bounds__(128) void k1_proj(
    const float* __restrict__ emb, const float* __restrict__ mask,
    const float* __restrict__ Bc, const float* __restrict__ Ac,
    float* __restrict__ U)
{
    __shared__ float sB[D_N * RD_N];     // 60 KB: B_c staged
    __shared__ float sY[CHUNK][18];      // Y tiles (pad 18 -> conflict-free)
    __shared__ float sAm[CHUNK][RL_N];   // A_c rows pre-scaled by mask

    const int b     = blockIdx.y;
    const int chunk = blockIdx.x;
    const int tid   = threadIdx.x;
    const int l0    = chunk * CHUNK;

    // stage B_c (960x16) into LDS, coalesced 16B loads
    {
        const float4* Bc4 = (const float4*)Bc;
        float4* sB4 = (float4*)sB;
        for (int i = tid; i < (D_N * RD_N) / 4; i += 128) sB4[i] = Bc4[i];
    }
    // stage A_c chunk with mask folded in: sAm[row][p] = A_c[l0+row][p]*mask[b][l0+row]
    for (int i = tid; i < CHUNK * RL_N; i += 128) {
        const int rr = i >> 4, p = i & 15;
        sAm[rr][p] = Ac[(size_t)(l0 + rr) * RL_N + p] *
                     mask[(size_t)b * L_N + l0 + rr];
    }
    __syncthreads();

    const int wave = tid >> 5, lane = tid & 31;
    const int m = lane & 15;     // row within 16-tile (A) / column n (B)
    const int h = lane >> 4;     // K-pair selector per ISA 16x4 f32 layout
    const float* Xrow = emb + ((size_t)b * L_N + l0 + wave * 16 + m) * D_N;

    v8f acc = {0.f, 0.f, 0.f, 0.f, 0.f, 0.f, 0.f, 0.f};
    #pragma unroll 4
    for (int k = 0; k < KSTEPS; ++k) {
        const int kc = 4 * k + 2 * h;
        // A fragment: lane holds X[m][kc], X[m][kc+1]  (8B aligned: kc even)
        v2f a = *(const v2f*)(Xrow + kc);
        // B fragment: lane holds B_c[kc][m], B_c[kc+1][m]
        v2f bb;
        bb.x = sB[kc * RD_N + m];
        bb.y = sB[(kc + 1) * RD_N + m];
        acc = __builtin_amdgcn_wmma_f32_16x16x4_f32(
            false, a, false, bb, (short)0, acc, false, false);
    }

    // C/D layout: VGPR i -> row (i + 8*h) of this wave's tile, col m
    #pragma unroll
    for (int i = 0; i < 8; ++i) sY[wave * 16 + i + 8 * h][m] = acc[i];
    __syncthreads();

    // partial U[p][r] = sum_rr sAm[rr][p] * sY[rr][r]; accumulate globally
    for (int t = tid; t < RL_N * RD_N; t += 128) {
        const int p = t >> 4, r = t & 15;
        float s = 0.f;
        #pragma unroll 8
        for (int rr = 0; rr < CHUNK; ++rr) s += sAm[rr][p] * sY[rr][r];
        atomicAdd(&U[b * (RL_N * RD_N) + t], s);
    }
}

// ---------------------------------------------------------------------------
__global__ void k_zero(float* __restrict__ p, int n)
{
    const int i = blockIdx.x * blockDim.x + threadIdx.x;
    if (i < n) p[i] = 0.f;
}

// per-row: z = U[i] @ H_c ; h[i] = z @ W1 + b1
__global__ __launch_bounds__(256) void k2_expand(
    const float* __restrict__ U, const float* __restrict__ Hc,
    const float* __restrict__ W1, const float* __restrict__ b1,
    float* __restrict__ hbuf)
{
    __shared__ float sU[RL_N * RD_N];
    __shared__ float sz[DLAT_N];
    const int i = blockIdx.x, tid = threadIdx.x;
    sU[tid] = U[i * (RL_N * RD_N) + tid];
    __syncthreads();
    for (int jj = tid; jj < DLAT_N; jj += 256) {
        float s = 0.f;
        #pragma unroll 4
        for (int k = 0; k < RL_N * RD_N; ++k) s += sU[k] * Hc[k * DLAT_N + jj];
        sz[jj] = s;
    }
    __syncthreads();
    for (int jj = tid; jj < DLAT_N; jj += 256) {
        float s = b1[jj];
        #pragma unroll 4
        for (int k = 0; k < DLAT_N; ++k) s += sz[k] * W1[k * DLAT_N + jj];
        hbuf[i * DLAT_N + jj] = s;
    }
}

// batch statistics (biased variance, like torch BN training)
__global__ __launch_bounds__(512) void k2_stats(
    const float* __restrict__ hbuf, float* __restrict__ mean,
    float* __restrict__ istd)
{
    const int j = threadIdx.x;
    float s = 0.f, s2 = 0.f;
    #pragma unroll 4
    for (int i = 0; i < B_N; ++i) {
        const float v = hbuf[i * DLAT_N + j];
        s += v; s2 += v * v;
    }
    const float mu  = s * (1.f / B_N);
    const float var = s2 * (1.f / B_N) - mu * mu;
    mean[j] = mu;
    istd[j] = rsqrtf(var + 1e-5f);
}

// per-row: out = relu(BN(h)) @ W2 + b2
__global__ __launch_bounds__(256) void k2_out(
    const float* __restrict__ hbuf, const float* __restrict__ mean,
    const float* __restrict__ istd, const float* __restrict__ gamma,
    const float* __restrict__ beta, const float* __restrict__ W2,
    const float* __restrict__ b2, float* __restrict__ out)
{
    __shared__ float sh[DLAT_N];
    const int i = blockIdx.x, tid = threadIdx.x;
    for (int jj = tid; jj < DLAT_N; jj += 256) {
        float v = (hbuf[i * DLAT_N + jj] - mean[jj]) * istd[jj] * gamma[jj] + beta[jj];
        sh[jj] = v > 0.f ? v : 0.f;
    }
    __syncthreads();
    for (int jj = tid; jj < DLAT_N; jj += 256) {
        float s = b2[jj];
        #pragma unroll 4
        for (int k = 0; k < DLAT_N; ++k) s += sh[k] * W2[k * DLAT_N + jj];
        out[i * DLAT_N + jj] = s;
    }
}

// ---------------------------------------------------------------------------
extern "C" void kernel_launch(void* const* d_in, const int* in_sizes, int n_in,
                              void* d_out, int out_size, void* d_ws, size_t ws_size,
                              hipStream_t stream)
{
    const float* emb   = (const float*)d_in[0];
    const float* mask  = (const float*)d_in[1];
    const float* Bc    = (const float*)d_in[2];
    const float* Ac    = (const float*)d_in[3];
    const float* Hc    = (const float*)d_in[4];
    const float* W1    = (const float*)d_in[5];
    const float* b1    = (const float*)d_in[6];
    const float* gamma = (const float*)d_in[7];
    const float* beta  = (const float*)d_in[8];
    const float* W2    = (const float*)d_in[9];
    const float* b2    = (const float*)d_in[10];
    float* out = (float*)d_out;

    char* ws = (char*)d_ws;
    float* U    = (float*)(ws);                              // 64*256 f32
    float* hbuf = (float*)(ws + 65536);                      // 64*512 f32
    float* mean = (float*)(ws + 65536 + 131072);             // 512 f32
    float* istd = (float*)(ws + 65536 + 131072 + 2048);      // 512 f32

    k_zero<<<dim3((B_N * RL_N * RD_N + 255) / 256), 256, 0, stream>>>(U, B_N * RL_N * RD_N);
    k1_proj<<<dim3(L_N / CHUNK, B_N), 128, 0, stream>>>(emb, mask, Bc, Ac, U);
    k2_expand<<<B_N, 256, 0, stream>>>(U, Hc, W1, b1, hbuf);
    k2_stats<<<1, DLAT_N, 0, stream>>>(hbuf, mean, istd);
    k2_out<<<B_N, 256, 0, stream>>>(hbuf, mean, istd, gamma, beta, W2, b2, out);
}